// Type12_50766513438939
// MI455X (gfx1250) — compile-verified
//
#include <hip/hip_runtime.h>
#include <math.h>
#include <stdint.h>

typedef float v2f __attribute__((ext_vector_type(2)));
typedef float v8f __attribute__((ext_vector_type(8)));

#define FAN_IN   512
#define FAN_MID  256
#define FAN_OUT  64
#define NEG_SLOPE 0.01f

#define LDA1 516   // 512 + 4 pad; 516*4 B row stride: 16B-multiple, bank-conflict free
#define LDA2 260   // 256 + 4 pad; 260*4 B row stride: 16B-multiple

__global__ void zero_f32(float* __restrict__ p, long long n) {
  long long i = (long long)blockIdx.x * blockDim.x + threadIdx.x;
  long long stride = (long long)gridDim.x * blockDim.x;
  for (; i < n; i += stride) p[i] = 0.0f;
}

// ---------------------------------------------------------------------------
// GEMM1: sup1[M,256] = x[M,512] @ W1[512,256], fp32 WMMA 16x16x4.
// Block = 512 threads = 16 waves. 16-row A stripe staged via CDNA5 async
// global->LDS (b128, ASYNCcnt); wave w computes the 16x16 tile at cols 16w.
// ---------------------------------------------------------------------------
__global__ __launch_bounds__(512) void gemm1_wmma(
    const float* __restrict__ x, const float* __restrict__ W1,
    float* __restrict__ sup, int M) {
  __shared__ __align__(16) float As[16 * LDA1];
  const int row0 = blockIdx.x * 16;

  // Async-stage A stripe: 16 x 512 f32 = 2048 float4 -> 4 b128 ops/thread.
  for (int idx = threadIdx.x; idx < 16 * (FAN_IN / 4); idx += 512) {
    int r  = idx >> 7;
    int c4 = (idx & 127) * 4;
    int gr = row0 + r;
    const float* gp = x + (size_t)(gr < M ? gr : 0) * FAN_IN + c4;  // tail-safe
    uint32_t lp = (uint32_t)(uintptr_t)&As[r * LDA1 + c4];  // flat[31:0] == LDS addr
    uint64_t ga = (uint64_t)(uintptr_t)gp;
    asm volatile("global_load_async_to_lds_b128 %0, %1, off"
                 :: "v"(lp), "v"(ga) : "memory");
  }
  asm volatile("s_wait_asynccnt 0x0" ::: "memory");
  __syncthreads();

  const int lane = threadIdx.x & 31;
  const int half = lane >> 4;      // 0: K=0,1   1: K=2,3  (A frag per ISA 7.12.2)
  const int mn   = lane & 15;      // A row (m) and B col (n) carried by this lane
  const int col0 = (threadIdx.x >> 5) * 16;

  v8f c = {};
#pragma unroll 4
  for (int k = 0; k < FAN_IN; k += 4) {
    const int kk = k + 2 * half;
    v2f a = *(const v2f*)&As[mn * LDA1 + kk];              // ds_load_b64
    v2f b;
    b.x = W1[(size_t)kk * FAN_MID + col0 + mn];            // L2-hot (512 KB)
    b.y = W1[(size_t)(kk + 1) * FAN_MID + col0 + mn];
    c = __builtin_amdgcn_wmma_f32_16x16x4_f32(false, a, false, b,
                                              (short)0, c, false, false);
  }

  float* outp = sup + (size_t)row0 * FAN_MID + col0 + mn;
  if (row0 + 16 <= M) {                                    // fast path: no exec juggling
#pragma unroll
    for (int j = 0; j < 8; ++j)
      outp[(size_t)(j + 8 * half) * FAN_MID] = c[j];
  } else {
#pragma unroll
    for (int j = 0; j < 8; ++j) {
      int m = row0 + j + 8 * half;
      if (m < M) sup[(size_t)m * FAN_MID + col0 + mn] = c[j];
    }
  }
}

// ---------------------------------------------------------------------------
// GEMM2: sup2[M,64] = leakyrelu(agg1 + b1) @ W2[256,64].
// Bias + activation fused into a float4-vectorized LDS stage. 128 thr = 4 waves.
// ---------------------------------------------------------------------------
__global__ __launch_bounds__(128) void gemm2_wmma(
    const float* __restrict__ agg1, const float* __restrict__ b1,
    const float* __restrict__ W2, float* __restrict__ sup2, int M) {
  __shared__ __align__(16) float As[16 * LDA2];
  const int row0 = blockIdx.x * 16;

  for (int idx = threadIdx.x; idx < 16 * (FAN_MID / 4); idx += 128) {
    int r  = idx >> 6;
    int c4 = (idx & 63) * 4;
    int gr = row0 + r;
    float4 v = make_float4(0.f, 0.f, 0.f, 0.f);
    if (gr < M) {
      v = *(const float4*)(agg1 + (size_t)gr * FAN_MID + c4);
      float4 bb = *(const float4*)(b1 + c4);
      v.x += bb.x; v.y += bb.y; v.z += bb.z; v.w += bb.w;
      v.x = (v.x > 0.f) ? v.x : v.x * NEG_SLOPE;           // leaky_relu fused
      v.y = (v.y > 0.f) ? v.y : v.y * NEG_SLOPE;
      v.z = (v.z > 0.f) ? v.z : v.z * NEG_SLOPE;
      v.w = (v.w > 0.f) ? v.w : v.w * NEG_SLOPE;
    }
    *(float4*)&As[r * LDA2 + c4] = v;                      // ds_store_b128
  }
  __syncthreads();

  const int lane = threadIdx.x & 31;
  const int half = lane >> 4;
  const int mn   = lane & 15;
  const int col0 = (threadIdx.x >> 5) * 16;

  v8f c = {};
#pragma unroll 4
  for (int k = 0; k < FAN_MID; k += 4) {
    const int kk = k + 2 * half;
    v2f a = *(const v2f*)&As[mn * LDA2 + kk];
    v2f b;
    b.x = W2[(size_t)kk * FAN_OUT + col0 + mn];            // W2 = 64 KB, L2-hot
    b.y = W2[(size_t)(kk + 1) * FAN_OUT + col0 + mn];
    c = __builtin_amdgcn_wmma_f32_16x16x4_f32(false, a, false, b,
                                              (short)0, c, false, false);
  }

  float* outp = sup2 + (size_t)row0 * FAN_OUT + col0 + mn;
  if (row0 + 16 <= M) {
#pragma unroll
    for (int j = 0; j < 8; ++j)
      outp[(size_t)(j + 8 * half) * FAN_OUT] = c[j];
  } else {
#pragma unroll
    for (int j = 0; j < 8; ++j) {
      int m = row0 + j + 8 * half;
      if (m < M) sup2[(size_t)m * FAN_OUT + col0 + mn] = c[j];
    }
  }
}

// ---------------------------------------------------------------------------
// Edge aggregation, F=256: wave per edge, lane covers 8 contiguous floats.
// Gather hits L2 (sup1 = 102 MB < 192 MB L2); scatter via hw f32 atomics.
// ---------------------------------------------------------------------------
__global__ __launch_bounds__(256) void edge_agg256(
    const float* __restrict__ sup, const int* __restrict__ src,
    const int* __restrict__ dst, const float* __restrict__ ew,
    float* __restrict__ agg, int nE) {
  int gid = blockIdx.x * 256 + threadIdx.x;
  int e = gid >> 5;
  if (e >= nE) return;
  int lane = gid & 31;
  int s = src[e], d = dst[e];
  float w = ew[e];
  const float4* srow = (const float4*)(sup + (size_t)s * FAN_MID) + lane * 2;
  float4 v0 = srow[0];
  float4 v1 = srow[1];
  float* arow = agg + (size_t)d * FAN_MID + lane * 8;
  __hip_atomic_fetch_add(arow + 0, v0.x * w, __ATOMIC_RELAXED, __HIP_MEMORY_SCOPE_AGENT);
  __hip_atomic_fetch_add(arow + 1, v0.y * w, __ATOMIC_RELAXED, __HIP_MEMORY_SCOPE_AGENT);
  __hip_atomic_fetch_add(arow + 2, v0.z * w, __ATOMIC_RELAXED, __HIP_MEMORY_SCOPE_AGENT);
  __hip_atomic_fetch_add(arow + 3, v0.w * w, __ATOMIC_RELAXED, __HIP_MEMORY_SCOPE_AGENT);
  __hip_atomic_fetch_add(arow + 4, v1.x * w, __ATOMIC_RELAXED, __HIP_MEMORY_SCOPE_AGENT);
  __hip_atomic_fetch_add(arow + 5, v1.y * w, __ATOMIC_RELAXED, __HIP_MEMORY_SCOPE_AGENT);
  __hip_atomic_fetch_add(arow + 6, v1.z * w, __ATOMIC_RELAXED, __HIP_MEMORY_SCOPE_AGENT);
  __hip_atomic_fetch_add(arow + 7, v1.w * w, __ATOMIC_RELAXED, __HIP_MEMORY_SCOPE_AGENT);
}

// Edge aggregation, F=64: wave per edge, lane covers 2 floats.
__global__ __launch_bounds__(256) void edge_agg64(
    const float* __restrict__ sup, const int* __restrict__ src,
    const int* __restrict__ dst, const float* __restrict__ ew,
    float* __restrict__ agg, int nE) {
  int gid = blockIdx.x * 256 + threadIdx.x;
  int e = gid >> 5;
  if (e >= nE) return;
  int lane = gid & 31;
  int s = src[e], d = dst[e];
  float w = ew[e];
  float2 v = *(const float2*)(sup + (size_t)s * FAN_OUT + lane * 2);
  float* arow = agg + (size_t)d * FAN_OUT + lane * 2;
  __hip_atomic_fetch_add(arow + 0, v.x * w, __ATOMIC_RELAXED, __HIP_MEMORY_SCOPE_AGENT);
  __hip_atomic_fetch_add(arow + 1, v.y * w, __ATOMIC_RELAXED, __HIP_MEMORY_SCOPE_AGENT);
}

// ---------------------------------------------------------------------------
// out = log_softmax(agg2 + b2) row-wise over 64 cols. One wave32 per row,
// 2 values per lane, shfl_xor tree reductions.
// ---------------------------------------------------------------------------
__global__ __launch_bounds__(256) void logsoftmax64(
    const float* __restrict__ agg2, const float* __restrict__ b2,
    float* __restrict__ out, int M) {
  int gid = blockIdx.x * 256 + threadIdx.x;
  int row = gid >> 5;
  if (row >= M) return;
  int lane = gid & 31;
  float2 v = *(const float2*)(agg2 + (size_t)row * FAN_OUT + lane * 2);
  v.x += b2[lane * 2];
  v.y += b2[lane * 2 + 1];
  float mx = fmaxf(v.x, v.y);
#pragma unroll
  for (int off = 16; off; off >>= 1) mx = fmaxf(mx, __shfl_xor(mx, off, 32));
  float s = __expf(v.x - mx) + __expf(v.y - mx);
#pragma unroll
  for (int off = 16; off; off >>= 1) s += __shfl_xor(s, off, 32);
  float lse = mx + __logf(s);
  float2 o;
  o.x = v.x - lse;
  o.y = v.y - lse;
  *(float2*)(out + (size_t)row * FAN_OUT + lane * 2) = o;
}

// ---------------------------------------------------------------------------
extern "C" void kernel_launch(void* const* d_in, const int* in_sizes, int n_in,
                              void* d_out, int out_size, void* d_ws, size_t ws_size,
                              hipStream_t stream) {
  const float* x    = (const float*)d_in[0];
  const int*   esrc = (const int*)  d_in[1];
  const int*   edst = (const int*)  d_in[2];
  const float* ew1  = (const float*)d_in[3];
  const float* ew2  = (const float*)d_in[4];
  const float* W1   = (const float*)d_in[5];
  const float* b1   = (const float*)d_in[6];
  const float* W2   = (const float*)d_in[7];
  const float* b2   = (const float*)d_in[8];
  float* out = (float*)d_out;

  const int M  = in_sizes[0] / FAN_IN;   // 100000
  const int nE = in_sizes[1];            // 3200000

  // Workspace layout (peak 2*M*256*4 = 204.8 MB):
  //   [0, M*256)            sup1   -- later reused for sup2 (M*64) ...
  //   [M*64, 2*M*64)        agg2   -- ... and agg2, both inside old sup1 region
  //   [M*256, 2*M*256)      agg1
  char* ws = (char*)d_ws;
  float* sup1 = (float*)ws;
  float* agg1 = (float*)(ws + (size_t)M * FAN_MID * sizeof(float));
  float* sup2 = (float*)ws;                                         // alias (sup1 dead)
  float* agg2 = (float*)(ws + (size_t)M * FAN_OUT * sizeof(float)); // alias (sup1 dead)

  const int mBlocks = (M + 15) / 16;

  // 1) agg1 = 0 (must re-init every launch; harness does not re-poison)
  zero_f32<<<2048, 256, 0, stream>>>(agg1, (long long)M * FAN_MID);
  // 2) sup1 = x @ W1
  gemm1_wmma<<<mBlocks, 512, 0, stream>>>(x, W1, sup1, M);
  // 3) agg1[dst] += sup1[src] * ew1
  edge_agg256<<<(nE * 32 + 255) / 256, 256, 0, stream>>>(sup1, esrc, edst, ew1, agg1, nE);
  // 4) agg2 = 0 (region free now that edge1 is done; stream-serialized)
  zero_f32<<<2048, 256, 0, stream>>>(agg2, (long long)M * FAN_OUT);
  // 5) sup2 = leakyrelu(agg1 + b1) @ W2   (activation fused in LDS stage)
  gemm2_wmma<<<mBlocks, 128, 0, stream>>>(agg1, b1, W2, sup2, M);
  // 6) agg2[dst] += sup2[src] * ew2
  edge_agg64<<<(nE * 32 + 255) / 256, 256, 0, stream>>>(sup2, esrc, edst, ew2, agg2, nE);
  // 7) out = log_softmax(agg2 + b2)
  logsoftmax64<<<(M * 32 + 255) / 256, 256, 0, stream>>>(agg2, b2, out, M);
}